// ConvAttention_89910845374839
// MI455X (gfx1250) — compile-verified
//
#include <hip/hip_runtime.h>

typedef float v2f __attribute__((ext_vector_type(2)));
typedef float v8f __attribute__((ext_vector_type(8)));

#define HIDDEN 128

// ---------------------------------------------------------------------------
// Zero the output accumulator (d_out is poisoned by the harness).
// ---------------------------------------------------------------------------
__global__ void convattn_zero_kernel(float* __restrict__ out, int n) {
    int i = blockIdx.x * blockDim.x + threadIdx.x;
    if (i < n) out[i] = 0.0f;
}

// ---------------------------------------------------------------------------
// q/k/v projections with V_WMMA_F32_16X16X4_F32.
// One wave per (16-node x 16-channel) tile; computes q, k, v tiles together,
// reusing the A (x) fragment across the three WMMAs per K-step.
//
// Layouts (CDNA5 ISA 7.12.2, 32-bit, wave32):
//   A 16x4 : lane l, vgpr v  ->  A[l&15][kb + 2*(l>>4) + v]      (float2 load)
//   B 4x16 : lane l, vgpr v  ->  B[kb + 2*(l>>4) + v][l&15]
//            here B = W.T, so element = W[chan = l&15][kb + 2*(l>>4) + v]
//   D 16x16: vgpr r          ->  D[node r + 8*(l>>4)][chan l&15]
// ---------------------------------------------------------------------------
__global__ __launch_bounds__(256) void convattn_qkv_wmma_kernel(
    const float* __restrict__ x,
    const float* __restrict__ Wq,
    const float* __restrict__ Wk,
    const float* __restrict__ Wv,
    float* __restrict__ q,
    float* __restrict__ k,
    float* __restrict__ v,
    int nNodes, int nodeTiles) {

    const int waveId   = blockIdx.x * (blockDim.x >> 5) + (threadIdx.x >> 5);
    const int lane     = threadIdx.x & 31;
    const int outTile  = waveId & 7;        // 128 / 16 = 8 channel tiles
    const int nodeTile = waveId >> 3;
    if (nodeTile >= nodeTiles) return;

    const int rc   = lane & 15;             // M index for A, N index for B
    const int half = lane >> 4;

    int mrow = nodeTile * 16 + rc;
    if (mrow >= nNodes) mrow = nNodes - 1;  // clamp tail: EXEC stays all-1s for WMMA

    const int chan = outTile * 16 + rc;

    const float* xp  = x  + (size_t)mrow * HIDDEN + 2 * half;
    const float* bqp = Wq + (size_t)chan * HIDDEN + 2 * half;
    const float* bkp = Wk + (size_t)chan * HIDDEN + 2 * half;
    const float* bvp = Wv + (size_t)chan * HIDDEN + 2 * half;

    v8f accq = {};
    v8f acck = {};
    v8f accv = {};

#pragma unroll 8
    for (int kb = 0; kb < HIDDEN; kb += 4) {
        v2f a  = *(const v2f*)(xp  + kb);
        v2f bq = *(const v2f*)(bqp + kb);
        v2f bk = *(const v2f*)(bkp + kb);
        v2f bv = *(const v2f*)(bvp + kb);
        accq = __builtin_amdgcn_wmma_f32_16x16x4_f32(false, a, false, bq, (short)0, accq, false, false);
        acck = __builtin_amdgcn_wmma_f32_16x16x4_f32(false, a, false, bk, (short)0, acck, false, false);
        accv = __builtin_amdgcn_wmma_f32_16x16x4_f32(false, a, false, bv, (short)0, accv, false, false);
    }

    // Store D tiles. Fold alpha's 1/sqrt(HEAD_DIM) = 0.25 into q.
#pragma unroll
    for (int r = 0; r < 8; ++r) {
        const int node = nodeTile * 16 + r + 8 * half;
        if (node < nNodes) {
            const size_t off = (size_t)node * HIDDEN + chan;
            q[off] = accq[r] * 0.25f;
            k[off] = acck[r];
            v[off] = accv[r];
        }
    }
}

// ---------------------------------------------------------------------------
// Edge phase: one wave32 per edge, lane l owns channels 4l..4l+3 (float4).
// Head h = channels 16h..16h+15 = lanes 4h..4h+3; per-head dot product
// reduced with shfl_xor(1), shfl_xor(2) inside each 4-lane group.
// q already carries the 0.25 scale.  out[dst] accumulated with f32 atomics.
// ---------------------------------------------------------------------------
__global__ __launch_bounds__(256) void convattn_edge_kernel(
    const float* __restrict__ q,
    const float* __restrict__ k,
    const float* __restrict__ v,
    const float* __restrict__ w_ij,
    const float* __restrict__ cutoff,
    const int*   __restrict__ src,   // edge_index[0] = j
    const int*   __restrict__ dst,   // edge_index[1] = i
    float* __restrict__ out,
    int nEdges) {

    const int gid  = blockIdx.x * blockDim.x + threadIdx.x;
    const int e    = gid >> 5;
    const int lane = threadIdx.x & 31;
    if (e >= nEdges) return;

    const int j = src[e];
    const int i = dst[e];
    const int c = lane * 4;

    const float4 qv = *(const float4*)(q + (size_t)i * HIDDEN + c);
    const float4 wv = *(const float4*)(w_ij + (size_t)e * HIDDEN + c);
    const float4 kv = *(const float4*)(k + (size_t)j * HIDDEN + c);
    const float4 vv = *(const float4*)(v + (size_t)j * HIDDEN + c);

    float p = qv.x * wv.x * kv.x + qv.y * wv.y * kv.y
            + qv.z * wv.z * kv.z + qv.w * wv.w * kv.w;
    // reduce over the 4 lanes of this head
    p += __shfl_xor(p, 1);
    p += __shfl_xor(p, 2);

    const float alpha = p * cutoff[e];

    float* o = out + (size_t)i * HIDDEN + c;
    atomicAdd(o + 0, alpha * vv.x);
    atomicAdd(o + 1, alpha * vv.y);
    atomicAdd(o + 2, alpha * vv.z);
    atomicAdd(o + 3, alpha * vv.w);
}

// ---------------------------------------------------------------------------
// Launch
// ---------------------------------------------------------------------------
extern "C" void kernel_launch(void* const* d_in, const int* in_sizes, int n_in,
                              void* d_out, int out_size, void* d_ws, size_t ws_size,
                              hipStream_t stream) {
    const float* x      = (const float*)d_in[0];
    const float* w_ij   = (const float*)d_in[1];
    const int*   eidx   = (const int*)  d_in[2];
    const float* cutoff = (const float*)d_in[3];
    const float* Wq     = (const float*)d_in[4];
    const float* Wk     = (const float*)d_in[5];
    const float* Wv     = (const float*)d_in[6];
    float* out          = (float*)d_out;

    const int nNodes = in_sizes[0] / HIDDEN;
    const int nEdges = in_sizes[3];          // cutoff is [E,1]
    const int* src = eidx;                   // edge_index[0]
    const int* dst = eidx + nEdges;          // edge_index[1]

    float* q = (float*)d_ws;
    float* k = q + (size_t)nNodes * HIDDEN;
    float* v = k + (size_t)nNodes * HIDDEN;

    // 1) zero the output accumulator
    const int outN = nNodes * HIDDEN;
    convattn_zero_kernel<<<(outN + 255) / 256, 256, 0, stream>>>(out, outN);

    // 2) q/k/v projections via fp32 WMMA
    const int nodeTiles  = (nNodes + 15) / 16;
    const int totalWaves = nodeTiles * 8;                // 8 channel tiles each
    const int gBlocks    = (totalWaves + 7) / 8;         // 8 waves per 256-thread block
    convattn_qkv_wmma_kernel<<<gBlocks, 256, 0, stream>>>(x, Wq, Wk, Wv, q, k, v,
                                                          nNodes, nodeTiles);

    // 3) edge attention + scatter-add
    const long long eThreads = (long long)nEdges * 32;
    const int eBlocks = (int)((eThreads + 255) / 256);
    convattn_edge_kernel<<<eBlocks, 256, 0, stream>>>(q, k, v, w_ij, cutoff,
                                                      src, dst, out, nEdges);
}